// Decoder_76476187672662
// MI455X (gfx1250) — compile-verified
//
#include <hip/hip_runtime.h>
#include <hip/hip_bf16.h>
#include <math.h>

// Problem dims (fixed by reference)
static constexpr int BB  = 512;   // batch
static constexpr int NN  = 128;   // nodes / steps
static constexpr int EE  = 256;   // embed
static constexpr int HH  = 256;   // hidden
static constexpr int CO  = 11;    // COU
static constexpr float C_TANH = 10.0f;

typedef float v2f __attribute__((ext_vector_type(2)));
typedef float v8f __attribute__((ext_vector_type(8)));

// gfx1250 has a hardware v_tanh_f32 (TRANS op, co-executes with VALU).
// Use the builtin when declared, otherwise force the instruction via asm.
__device__ __forceinline__ float fast_tanh(float x) {
#if __has_builtin(__builtin_amdgcn_tanhf)
  return __builtin_amdgcn_tanhf(x);
#else
  float r;
  asm volatile("v_tanh_f32 %0, %1" : "=v"(r) : "v"(x));
  return r;
#endif
}
__device__ __forceinline__ float sigmoidf(float x) {
  return 1.0f / (1.0f + __expf(-x));
}

// ---------------------------------------------------------------------------
// Generic fp32 WMMA GEMM:  Y[m][n] (+)= sum_k X[m][k] * W[n][k]  + b0[n]+b1[n]
// One wave (32 threads) per 16x32 output tile (two 16x16 WMMA accumulators,
// A fragment reused for both), K looped in steps of 4 using
// V_WMMA_F32_16X16X4_F32.  blockIdx.z slices X (stride xz) and offsets Y
// (stride yz) so the same kernel computes e = einsum('oh,lbh->bol').
// ---------------------------------------------------------------------------
__global__ void wmma_gemm_nt(const float* __restrict__ X, int ldx, long xz,
                             const float* __restrict__ W, int ldw,
                             const float* __restrict__ b0,
                             const float* __restrict__ b1,
                             float* __restrict__ Y, long ldy, int ynstride,
                             long yz, int K, int accum)
{
  const float* Xp = X + (size_t)blockIdx.z * (size_t)xz;
  float*       Yp = Y + (size_t)blockIdx.z * (size_t)yz;
  const int row0 = blockIdx.x * 16;
  const int col0 = blockIdx.y * 32;      // two adjacent 16-wide N tiles
  const int lane = threadIdx.x;          // 0..31, full wave (EXEC all ones)
  const int hv   = lane >> 4;            // half-wave select
  const int lm   = lane & 15;

  const float* xrow  = Xp + (size_t)(row0 + lm) * ldx;
  const float* wrow0 = W  + (size_t)(col0 + lm) * ldw;
  const float* wrow1 = W  + (size_t)(col0 + 16 + lm) * ldw;

  v8f acc0 = {0.f,0.f,0.f,0.f,0.f,0.f,0.f,0.f};
  v8f acc1 = {0.f,0.f,0.f,0.f,0.f,0.f,0.f,0.f};
  for (int k = 0; k < K; k += 4) {
    // A 16x4 fp32 layout: lanes 0-15 hold K={0,1}, lanes 16-31 hold K={2,3}
    v2f a, b0v, b1v;
    a.x   = xrow[k + 2*hv];    a.y   = xrow[k + 2*hv + 1];
    b0v.x = wrow0[k + 2*hv];   b0v.y = wrow0[k + 2*hv + 1];
    b1v.x = wrow1[k + 2*hv];   b1v.y = wrow1[k + 2*hv + 1];
    acc0 = __builtin_amdgcn_wmma_f32_16x16x4_f32(
        false, a, false, b0v, (short)0, acc0, false, false);
    acc1 = __builtin_amdgcn_wmma_f32_16x16x4_f32(
        false, a, false, b1v, (short)0, acc1, false, false);
  }

  const int n0 = col0 + lm;
  const int n1 = col0 + 16 + lm;
  float bias0 = 0.f, bias1 = 0.f;
  if (b0) { bias0 += b0[n0]; bias1 += b0[n1]; }
  if (b1) { bias0 += b1[n0]; bias1 += b1[n1]; }
#pragma unroll
  for (int r = 0; r < 8; ++r) {
    const int m = row0 + r + hv * 8;     // C/D: VGPR r -> M=r (+8 upper half)
    const size_t yi0 = (size_t)m * (size_t)ldy + (size_t)n0 * (size_t)ynstride;
    const size_t yi1 = (size_t)m * (size_t)ldy + (size_t)n1 * (size_t)ynstride;
    float v0 = acc0[r] + bias0;
    float v1 = acc1[r] + bias1;
    if (accum) { v0 += Yp[yi0]; v1 += Yp[yi1]; }
    Yp[yi0] = v0;
    Yp[yi1] = v1;
  }
}

// ---------------------------------------------------------------------------
// State init: copy carries into workspace (harness poisons ws with 0xAA).
// ---------------------------------------------------------------------------
__global__ void init_state_kernel(const float* __restrict__ dec0,
                                  const float* __restrict__ h0,
                                  const float* __restrict__ c0,
                                  const unsigned char* __restrict__ reach,
                                  const unsigned char* __restrict__ vdm0,
                                  float* __restrict__ dec, float* __restrict__ h,
                                  float* __restrict__ c,
                                  unsigned char* __restrict__ mask,
                                  unsigned char* __restrict__ vdm,
                                  int* __restrict__ idxs)
{
  const size_t i = (size_t)blockIdx.x * blockDim.x + threadIdx.x;
  if (i < (size_t)BB * NN * NN) vdm[i] = vdm0[i];
  if (i < (size_t)BB * HH) { dec[i] = dec0[i]; h[i] = h0[i]; c[i] = c0[i]; }
  if (i < (size_t)BB * NN) mask[i] = reach[i];
  if (i < (size_t)BB) idxs[i] = 0;
}

// ---------------------------------------------------------------------------
// LSTM elementwise update from gates (B x 4H, [i|f|g|o]).
// ---------------------------------------------------------------------------
__global__ void lstm_update_kernel(const float* __restrict__ gates,
                                   float* __restrict__ h, float* __restrict__ c)
{
  const int i = blockIdx.x * blockDim.x + threadIdx.x;
  if (i >= BB * HH) return;
  const int b = i / HH, k = i % HH;
  const float* g = gates + (size_t)b * 4 * HH;
  const float gi = g[k], gf = g[HH + k], gg = g[2*HH + k], go = g[3*HH + k];
  const float cn = sigmoidf(gf) * c[i] + sigmoidf(gi) * fast_tanh(gg);
  const float hn = sigmoidf(go) * fast_tanh(cn);
  c[i] = cn; h[i] = hn;
}

// merge tail: add the COU (=11) columns of W_merge (row stride H+CO).
__global__ void merge_tail_kernel(float* __restrict__ gl,
                                  const float* __restrict__ cou,
                                  const float* __restrict__ Wm)
{
  const int i = blockIdx.x * blockDim.x + threadIdx.x;
  if (i >= BB * HH) return;
  const int b = i / HH, o = i % HH;
  float acc = gl[i];
  const float* wr = Wm + (size_t)o * (HH + CO) + HH;
  const float* cr = cou + (size_t)b * CO;
#pragma unroll
  for (int j = 0; j < CO; ++j) acc += cr[j] * wr[j];
  gl[i] = acc;
}

// ---------------------------------------------------------------------------
// Per-step mask update: scatter idxs (for step>0), _mask_modify, full_mask.
// One block per batch row, 128 threads.
// ---------------------------------------------------------------------------
__global__ void mask_update_kernel(unsigned char* __restrict__ mask,
                                   const int* __restrict__ idxs,
                                   const unsigned char* __restrict__ vdm,
                                   unsigned char* __restrict__ fullmask,
                                   int step)
{
  __shared__ int red[NN];
  const int b = blockIdx.x, tid = threadIdx.x;
  if (step > 0 && tid == 0) mask[b * NN + idxs[b]] = 1;
  __syncthreads();
  red[tid] = mask[b * NN + tid] ? 1 : 0;
  __syncthreads();
  for (int s = 64; s > 0; s >>= 1) {
    if (tid < s) red[tid] &= red[tid + s];
    __syncthreads();
  }
  if (red[0] && tid == 0) mask[b * NN + NN - 1] = 0;   // _mask_modify
  __syncthreads();
  const unsigned char fm =
      (mask[b * NN + tid] | vdm[((size_t)b * NN + step) * NN + tid]) ? 1 : 0;
  fullmask[b * NN + tid] = fm;
}

// ---------------------------------------------------------------------------
// glimpse attention: logits -> masked softmax -> g_l2 = e_gl @ p.
// One block (128 threads = 4 waves) per batch row; e stays L2-resident.
// ---------------------------------------------------------------------------
__global__ void attn_gl_kernel(const float* __restrict__ q,
                               const float* __restrict__ e,
                               const float* __restrict__ v,
                               const unsigned char* __restrict__ fullmask,
                               float* __restrict__ gout)
{
  __shared__ float q_s[HH], v_s[HH], red[NN], p_s[NN];
  const int b = blockIdx.x, tid = threadIdx.x;
  q_s[tid] = q[b * HH + tid];        q_s[tid + NN] = q[b * HH + NN + tid];
  v_s[tid] = v[tid];                 v_s[tid + NN] = v[tid + NN];
  __syncthreads();

  const float* eb = e + (size_t)b * HH * NN;
  const float* ec = eb + tid;                  // column l=tid, stride NN
  float acc = 0.f;
#pragma unroll 8
  for (int hh = 0; hh < HH; ++hh)
    acc += v_s[hh] * fast_tanh(q_s[hh] + ec[(size_t)hh * NN]);

  const bool masked = fullmask[b * NN + tid] != 0;
  const float logit = masked ? -INFINITY : acc;

  red[tid] = logit; __syncthreads();
  for (int s = 64; s > 0; s >>= 1) { if (tid < s) red[tid] = fmaxf(red[tid], red[tid + s]); __syncthreads(); }
  const float mx = red[0]; __syncthreads();
  const float p = masked ? 0.f : __expf(logit - mx);
  red[tid] = p; __syncthreads();
  for (int s = 64; s > 0; s >>= 1) { if (tid < s) red[tid] += red[tid + s]; __syncthreads(); }
  const float inv = 1.0f / red[0];
  p_s[tid] = p * inv; __syncthreads();

  for (int hh = tid; hh < HH; hh += NN) {
    const float* er = eb + (size_t)hh * NN;
    float a2 = 0.f;
#pragma unroll 8
    for (int l = 0; l < NN; ++l) a2 += er[l] * p_s[l];
    gout[b * HH + hh] = a2;
  }
}

// ---------------------------------------------------------------------------
// pointer attention + log_softmax + argmax + state updates + outputs.
// ---------------------------------------------------------------------------
__global__ void attn_ptr_kernel(const float* __restrict__ q,
                                const float* __restrict__ e,
                                const float* __restrict__ v,
                                const unsigned char* __restrict__ fullmask,
                                const unsigned char* __restrict__ mask,
                                int* __restrict__ idxs,
                                const float* __restrict__ embedded,
                                const float* __restrict__ bmE,
                                unsigned char* __restrict__ vdm,
                                float* __restrict__ dec,
                                float* __restrict__ out, int step)
{
  __shared__ float q_s[HH], v_s[HH], red[NN];
  __shared__ int ri[NN], red_i[NN];
  const int b = blockIdx.x, tid = threadIdx.x;
  q_s[tid] = q[b * HH + tid];        q_s[tid + NN] = q[b * HH + NN + tid];
  v_s[tid] = v[tid];                 v_s[tid + NN] = v[tid + NN];
  __syncthreads();

  const float* eb = e + (size_t)b * HH * NN;
  const float* ec = eb + tid;
  float acc = 0.f;
#pragma unroll 8
  for (int hh = 0; hh < HH; ++hh)
    acc += v_s[hh] * fast_tanh(q_s[hh] + ec[(size_t)hh * NN]);

  const bool masked = fullmask[b * NN + tid] != 0;
  const float logit = masked ? -INFINITY : C_TANH * fast_tanh(acc);

  // log-softmax
  red[tid] = logit; __syncthreads();
  for (int s = 64; s > 0; s >>= 1) { if (tid < s) red[tid] = fmaxf(red[tid], red[tid + s]); __syncthreads(); }
  const float mx = red[0]; __syncthreads();
  const float ex = masked ? 0.f : __expf(logit - mx);
  red[tid] = ex; __syncthreads();
  for (int s = 64; s > 0; s >>= 1) { if (tid < s) red[tid] += red[tid + s]; __syncthreads(); }
  const float lse = logf(red[0]);
  const float logp = masked ? -INFINITY : (logit - mx - lse);
  out[((size_t)b * NN + step) * NN + tid] = logp;
  __syncthreads();

  // argmax (first max, like jnp.argmax)
  red[tid] = logp; ri[tid] = tid; __syncthreads();
  for (int s = 64; s > 0; s >>= 1) {
    if (tid < s) {
      if (red[tid + s] > red[tid] ||
          (red[tid + s] == red[tid] && ri[tid + s] < ri[tid])) {
        red[tid] = red[tid + s]; ri[tid] = ri[tid + s];
      }
    }
    __syncthreads();
  }
  const int new_idx = ri[0];
  __syncthreads();

  if (tid == 0) {
    idxs[b] = new_idx;
    out[(size_t)BB * NN * NN + (size_t)b * NN + step] = (float)new_idx;  // sels
  }
  // next decoder input: embedded_inputs[new_idx, b, :]
  const float* src = embedded + ((size_t)new_idx * BB + b) * EE;
  dec[b * EE + tid] = src[tid];
  dec[b * EE + NN + tid] = src[NN + tid];

  // V_decode_mask update
  const bool unv = (mask[b * NN + tid] == 0);
  red_i[tid] = unv ? 1 : 0;
  const float score = unv ? bmE[((size_t)b * NN + new_idx) * NN + tid] : 0.f;
  red[tid] = score; ri[tid] = tid; __syncthreads();
  for (int s = 64; s > 0; s >>= 1) {
    if (tid < s) {
      red_i[tid] += red_i[tid + s];
      if (red[tid + s] > red[tid] ||
          (red[tid + s] == red[tid] && ri[tid + s] < ri[tid])) {
        red[tid] = red[tid + s]; ri[tid] = ri[tid + s];
      }
    }
    __syncthreads();
  }
  if (tid == 0) {
    const bool valid = red_i[0] >= 3;
    int row = step + 1; if (row > NN - 1) row = NN - 1;
    if (valid) vdm[((size_t)b * NN + row) * NN + ri[0]] = 1;
  }
}

__global__ void final_mask_kernel(const unsigned char* __restrict__ mask,
                                  float* __restrict__ out)
{
  const int i = blockIdx.x * blockDim.x + threadIdx.x;
  if (i < BB * NN)
    out[(size_t)BB * NN * NN + (size_t)BB * NN + i] = mask[i] ? 1.0f : 0.0f;
}

// ---------------------------------------------------------------------------
extern "C" void kernel_launch(void* const* d_in, const int* in_sizes, int n_in,
                              void* d_out, int out_size, void* d_ws, size_t ws_size,
                              hipStream_t stream)
{
  (void)in_sizes; (void)n_in; (void)out_size; (void)ws_size;
  const float* decoder_input = (const float*)d_in[0];
  const float* embedded      = (const float*)d_in[1];
  const float* h0            = (const float*)d_in[2];
  const float* c0            = (const float*)d_in[3];
  const float* context       = (const float*)d_in[4];
  const unsigned char* reach = (const unsigned char*)d_in[5];
  const float* embed_cou     = (const float*)d_in[6];
  const unsigned char* vdm0  = (const unsigned char*)d_in[7];
  const float* bmE           = (const float*)d_in[8];
  const float* W_ih          = (const float*)d_in[9];
  const float* W_hh          = (const float*)d_in[10];
  const float* b_ih          = (const float*)d_in[11];
  const float* b_hh          = (const float*)d_in[12];
  const float* W_merge       = (const float*)d_in[13];
  const float* b_merge       = (const float*)d_in[14];
  const float* gl_Wq         = (const float*)d_in[15];
  const float* gl_bq         = (const float*)d_in[16];
  const float* gl_Wref       = (const float*)d_in[17];
  const float* gl_bref       = (const float*)d_in[18];
  const float* gl_v          = (const float*)d_in[19];
  const float* ptr_Wq        = (const float*)d_in[20];
  const float* ptr_bq        = (const float*)d_in[21];
  const float* ptr_Wref      = (const float*)d_in[22];
  const float* ptr_bref      = (const float*)d_in[23];
  const float* ptr_v         = (const float*)d_in[24];
  float* out = (float*)d_out;

  // workspace carve (256B aligned)
  char* wsb = (char*)d_ws;
  size_t off = 0;
  auto carve = [&](size_t bytes) {
    void* p = wsb + off;
    off = (off + bytes + 255) & ~(size_t)255;
    return p;
  };
  float* e_gl  = (float*)carve((size_t)BB * HH * NN * 4);   // 67 MB (L2-resident)
  float* e_ptr = (float*)carve((size_t)BB * HH * NN * 4);   // 67 MB
  float* h     = (float*)carve((size_t)BB * HH * 4);
  float* c     = (float*)carve((size_t)BB * HH * 4);
  float* dec   = (float*)carve((size_t)BB * EE * 4);
  float* gl    = (float*)carve((size_t)BB * HH * 4);        // g_l, then g_l2
  float* qbuf  = (float*)carve((size_t)BB * HH * 4);
  float* gates = (float*)carve((size_t)BB * 4 * HH * 4);
  unsigned char* mask     = (unsigned char*)carve((size_t)BB * NN);
  unsigned char* fullmask = (unsigned char*)carve((size_t)BB * NN);
  unsigned char* vdm      = (unsigned char*)carve((size_t)BB * NN * NN);
  int* idxs               = (int*)carve((size_t)BB * 4);

  // init carries
  init_state_kernel<<<(BB * NN * NN) / 256, 256, 0, stream>>>(
      decoder_input, h0, c0, reach, vdm0, dec, h, c, mask, vdm, idxs);

  // precompute e_gl / e_ptr: e[b][o][l] = sum_h Wref[o][h]*context[l][b][h] + bref[o]
  {
    dim3 g(BB / 16, HH / 32, NN);
    wmma_gemm_nt<<<g, 32, 0, stream>>>(context, HH, (long)BB * HH,
                                       gl_Wref, HH, gl_bref, nullptr,
                                       e_gl, (long)HH * NN, NN, 1L, HH, 0);
    wmma_gemm_nt<<<g, 32, 0, stream>>>(context, HH, (long)BB * HH,
                                       ptr_Wref, HH, ptr_bref, nullptr,
                                       e_ptr, (long)HH * NN, NN, 1L, HH, 0);
  }

  const dim3 gGates(BB / 16, (4 * HH) / 32);   // 32 x 32 tiles (16x32 each)
  const dim3 gH(BB / 16, HH / 32);             // 32 x 8 tiles

  for (int step = 0; step < NN; ++step) {
    // LSTM gates = dec@W_ih.T + b_ih + b_hh   (pass 1)
    wmma_gemm_nt<<<gGates, 32, 0, stream>>>(dec, EE, 0L, W_ih, EE, b_ih, b_hh,
                                            gates, 4 * HH, 1, 0L, EE, 0);
    //            += h@W_hh.T                  (pass 2, accumulate)
    wmma_gemm_nt<<<gGates, 32, 0, stream>>>(h, HH, 0L, W_hh, HH, nullptr, nullptr,
                                            gates, 4 * HH, 1, 0L, HH, 1);
    lstm_update_kernel<<<(BB * HH) / 256, 256, 0, stream>>>(gates, h, c);

    // g_l = [h_new | embed_cou] @ W_merge.T + b_merge
    wmma_gemm_nt<<<gH, 32, 0, stream>>>(h, HH, 0L, W_merge, HH + CO, b_merge,
                                        nullptr, gl, HH, 1, 0L, HH, 0);
    merge_tail_kernel<<<(BB * HH) / 256, 256, 0, stream>>>(gl, embed_cou, W_merge);

    // masks for this step
    mask_update_kernel<<<BB, NN, 0, stream>>>(mask, idxs, vdm, fullmask, step);

    // glimpse: q = g_l@Wq.T + bq ; softmax attention -> g_l2 (into gl)
    wmma_gemm_nt<<<gH, 32, 0, stream>>>(gl, HH, 0L, gl_Wq, HH, gl_bq, nullptr,
                                        qbuf, HH, 1, 0L, HH, 0);
    attn_gl_kernel<<<BB, NN, 0, stream>>>(qbuf, e_gl, gl_v, fullmask, gl);

    // pointer: q = g_l2@Wq.T + bq ; log_softmax, argmax, state updates
    wmma_gemm_nt<<<gH, 32, 0, stream>>>(gl, HH, 0L, ptr_Wq, HH, ptr_bq, nullptr,
                                        qbuf, HH, 1, 0L, HH, 0);
    attn_ptr_kernel<<<BB, NN, 0, stream>>>(qbuf, e_ptr, ptr_v, fullmask, mask,
                                           idxs, embedded, bmE, vdm, dec, out,
                                           step);
  }

  final_mask_kernel<<<(BB * NN) / 256, 256, 0, stream>>>(mask, out);
}